// GATTransformerEncoderLayer_8203387535545
// MI455X (gfx1250) — compile-verified
//
#include <hip/hip_runtime.h>
#include <hip/hip_bf16.h>

#define H 8
#define C 32
#define NEG_SLOPE 0.2f
#define LN_EPS 1e-5f

typedef float v2f __attribute__((ext_vector_type(2)));
typedef float v8f __attribute__((ext_vector_type(8)));

// ---------------------------------------------------------------------------
// FP32 WMMA: D = A(16x4) * B(4x16) + C(16x16), wave32.
// Operand layout (CDNA5 ISA §7.12.2, 32-bit A 16x4 table):
//   A: lane L (0-15): M=L, K={0,1} in {a.x,a.y}; lane L (16-31): M=L-16, K={2,3}
//   B: lane L (0-15): N=L, K={0,1};              lane L (16-31): N=L-16, K={2,3}
//   C/D: vgpr v: lanes 0-15 -> (M=v,   N=lane); lanes 16-31 -> (M=v+8, N=lane-16)
// ---------------------------------------------------------------------------
__device__ __forceinline__ v8f wmma_f32(v2f a, v2f b, v8f c) {
  return __builtin_amdgcn_wmma_f32_16x16x4_f32(
      /*neg_a=*/false, a, /*neg_b=*/false, b,
      /*c_mod=*/(short)0, c, /*reuse_a=*/false, /*reuse_b=*/false);
}

// ---------------------------------------------------------------------------
// Tensor Data Mover staging: one wave-scoped DMA (tracked by TENSORcnt)
// replaces the per-thread global->LDS copy loop. Descriptor encoding per
// CDNA5 ISA: group0 = {count=1, lds_addr,
// global_addr[56:0], type=2}; group1 = {data_size=4B, tensor_dim0=nelem,
// tensor_dim1=1, tile_dim0=nelem, tile_dim1=1, dim0_stride=nelem}.
// This toolchain exposes the 6-arg builtin:
//   (u32x4 g0, i32x8 g1, i32x4 g2, i32x4 g3, i32x8 extra, i32 cpol)
// Groups 2/3 NULL (<=2D tensor), extra zero-filled.
// ---------------------------------------------------------------------------
#if __has_builtin(__builtin_amdgcn_tensor_load_to_lds) && \
    __has_builtin(__builtin_amdgcn_s_wait_tensorcnt)
#define USE_TDM 1
typedef unsigned int u32x4 __attribute__((ext_vector_type(4)));
typedef int i32x4 __attribute__((ext_vector_type(4)));
typedef int i32x8 __attribute__((ext_vector_type(8)));

__device__ __forceinline__ void tdm_load_to_lds(const float* gsrc, void* ldst,
                                                unsigned nelem) {
  unsigned long long ga = (unsigned long long)(uintptr_t)gsrc;
  unsigned lds_off = (unsigned)(uintptr_t)ldst;  // low 32 bits = LDS offset
  u32x4 g0;
  g0.x = 0x1u;                                   // count=1 (valid descriptor)
  g0.y = lds_off;                                // lds_addr [63:32]
  g0.z = (unsigned)(ga & 0xFFFFFFFFull);         // global_addr[31:0]
  g0.w = (unsigned)((ga >> 32) & 0x01FFFFFFull)  // global_addr[56:32]
         | 0x80000000u;                          // type=2 at bits [127:126]
  i32x8 g1;
  g1[0] = 0x2 << 16;                        // wg_mask=0, data_size=2 (4 bytes)
  g1[1] = (int)((nelem & 0xFFFFu) << 16);   // tensor_dim0[15:0]  @ bits 63:48
  g1[2] = (int)((nelem >> 16) & 0xFFFFu)    // tensor_dim0[31:16] @ bits 79:64
          | (1 << 16);                      // tensor_dim1 = 1    @ bits 95:80
  g1[3] = (int)((nelem & 0xFFFFu) << 16);   // tile_dim0          @ bits 127:112
  g1[4] = 1;                                // tile_dim1 = 1, tile_dim2 = 0
  g1[5] = (int)nelem;                       // tensor_dim0_stride[31:0]
  g1[6] = 0;
  g1[7] = 0;
  i32x4 z4 = {0, 0, 0, 0};
  i32x8 z8 = {0, 0, 0, 0, 0, 0, 0, 0};
  __builtin_amdgcn_tensor_load_to_lds(g0, g1, z4, z4, z8, 0);
}
#endif

// Order-preserving float -> uint encoding for atomic max over signed floats.
__device__ __forceinline__ unsigned enc_f32(float f) {
  unsigned u = __float_as_uint(f);
  return (u & 0x80000000u) ? ~u : (u | 0x80000000u);
}
__device__ __forceinline__ float dec_f32(unsigned u) {
  return (u & 0x80000000u) ? __uint_as_float(u & 0x7FFFFFFFu)
                           : __uint_as_float(~u);
}

__device__ __forceinline__ float leaky(float x) {
  return x > 0.0f ? x : NEG_SLOPE * x;
}

// ---------------------------------------------------------------------------
// K0: initialize workspace accumulators (harness does not re-poison between
// replays, so we must re-init every call).
// ---------------------------------------------------------------------------
__global__ void k_init(unsigned* __restrict__ amax, float* __restrict__ denom,
                       float* __restrict__ accum, int n) {
  int i = blockIdx.x * blockDim.x + threadIdx.x;
  if (i < n * H) { amax[i] = 0u; denom[i] = 0.0f; }
  if (i < n * C) accum[i] = 0.0f;
}

// ---------------------------------------------------------------------------
// K1: xp = x @ W_gat   (N x 32) * (32 x 256) -> (N x 256), FP32 WMMA.
// Block = 256 threads = 8 waves; each wave owns a 16-node row tile.
// W_gat (32 KB) staged in LDS via TDM.  Requires n % 16 == 0 (n = 100000).
// ---------------------------------------------------------------------------
__global__ void __launch_bounds__(256) k_gemm_xp(
    const float* __restrict__ x, const float* __restrict__ W,
    float* __restrict__ xp, int n) {
  __shared__ float Wlds[32 * 256];
#ifdef USE_TDM
  if (threadIdx.x < 32) {
    tdm_load_to_lds(W, Wlds, 32 * 256);
    __builtin_amdgcn_s_wait_tensorcnt(0);
  }
#else
  for (int i = threadIdx.x; i < 32 * 256; i += blockDim.x) Wlds[i] = W[i];
#endif
  __syncthreads();

  const int wave = threadIdx.x >> 5;
  const int lane = threadIdx.x & 31;
  const int node0 = (blockIdx.x * 8 + wave) * 16;
  if (node0 >= n) return;

  const int row = lane & 15;
  const int khalf = (lane >> 4) * 2;  // 0 or 2
  const int col = lane & 15;
  const int half = lane >> 4;

  // Preload all 8 A k-step fragments for this 16x32 row tile.
  v2f a[8];
  const float* xr = x + (size_t)(node0 + row) * C;
#pragma unroll
  for (int kk = 0; kk < 8; ++kk) {
    int kb = kk * 4 + khalf;
    a[kk].x = xr[kb];
    a[kk].y = xr[kb + 1];
  }

#pragma unroll 1
  for (int t = 0; t < 16; ++t) {  // 16 column tiles of 16 -> 256 outputs
    v8f c = {};
#pragma unroll
    for (int kk = 0; kk < 8; ++kk) {
      int kr = kk * 4 + khalf;
      v2f b;
      b.x = Wlds[kr * 256 + t * 16 + col];
      b.y = Wlds[(kr + 1) * 256 + t * 16 + col];
      c = wmma_f32(a[kk], b, c);
    }
#pragma unroll
    for (int v = 0; v < 8; ++v) {
      int m = v + half * 8;
      xp[(size_t)(node0 + m) * 256 + t * 16 + col] = c[v];
    }
  }
}

// ---------------------------------------------------------------------------
// K2: per-(node, head) attention logits  a_src[n,h] = <xp[n,h,:], att_src[h,:]>
// ---------------------------------------------------------------------------
__global__ void k_att(const float* __restrict__ xp,
                      const float* __restrict__ att_src,
                      const float* __restrict__ att_dst,
                      float* __restrict__ a_src, float* __restrict__ a_dst,
                      int n) {
  int i = blockIdx.x * blockDim.x + threadIdx.x;
  if (i >= n * H) return;
  int h = i & (H - 1);
  int nd = i >> 3;
  const float* v = xp + (size_t)nd * 256 + h * C;
  const float* as = att_src + h * C;
  const float* ad = att_dst + h * C;
  float s = 0.0f, d = 0.0f;
#pragma unroll
  for (int c = 0; c < C; ++c) {
    float xv = v[c];
    s += xv * as[c];
    d += xv * ad[c];
  }
  a_src[i] = s;
  a_dst[i] = d;
}

// ---------------------------------------------------------------------------
// Edge passes over E' = E + N edges (last N are self loops, src = dst = e - E).
// Edge lists are streamed exactly once per pass -> non-temporal loads so the
// 25.6 MB streams do not evict the L2-resident xp (102.4 MB of 192 MB L2).
// ---------------------------------------------------------------------------
__device__ __forceinline__ void edge_nodes(int e, const long long* __restrict__ src,
                                           const long long* __restrict__ dst,
                                           int E, int& s, int& d) {
  if (e < E) {
    s = (int)__builtin_nontemporal_load(src + e);
    d = (int)__builtin_nontemporal_load(dst + e);
  } else {
    s = e - E;
    d = e - E;
  }
}

// K3: segment max (atomic max on order-preserving uint encoding).
__global__ void k_edge_max(const long long* __restrict__ src,
                           const long long* __restrict__ dst,
                           const float* __restrict__ a_src,
                           const float* __restrict__ a_dst,
                           unsigned* __restrict__ amax, int E, int n) {
  int e = blockIdx.x * blockDim.x + threadIdx.x;
  if (e >= E + n) return;
  int s, d;
  edge_nodes(e, src, dst, E, s, d);
#pragma unroll
  for (int h = 0; h < H; ++h) {
    float al = leaky(a_src[s * H + h] + a_dst[d * H + h]);
    atomicMax(&amax[d * H + h], enc_f32(al));
  }
}

// K4: segment sum of exp(alpha - max).
__global__ void k_edge_sum(const long long* __restrict__ src,
                           const long long* __restrict__ dst,
                           const float* __restrict__ a_src,
                           const float* __restrict__ a_dst,
                           const unsigned* __restrict__ amax,
                           float* __restrict__ denom, int E, int n) {
  int e = blockIdx.x * blockDim.x + threadIdx.x;
  if (e >= E + n) return;
  int s, d;
  edge_nodes(e, src, dst, E, s, d);
#pragma unroll
  for (int h = 0; h < H; ++h) {
    float al = leaky(a_src[s * H + h] + a_dst[d * H + h]);
    float ea = __expf(al - dec_f32(amax[d * H + h]));
    __hip_atomic_fetch_add(&denom[d * H + h], ea, __ATOMIC_RELAXED,
                           __HIP_MEMORY_SCOPE_AGENT);
  }
}

// K5: aggregation with the head-mean folded in:
//   accum[d, c] += (1/8) * sum_h xp[s, h, c] * attn[e, h]
// 32 f32 atomics per edge instead of 256; xp gather is L2-resident.
__global__ void k_edge_agg(const long long* __restrict__ src,
                           const long long* __restrict__ dst,
                           const float* __restrict__ a_src,
                           const float* __restrict__ a_dst,
                           const unsigned* __restrict__ amax,
                           const float* __restrict__ denom,
                           const float* __restrict__ xp,
                           float* __restrict__ accum, int E, int n) {
  int e = blockIdx.x * blockDim.x + threadIdx.x;
  if (e >= E + n) return;
  int s, d;
  edge_nodes(e, src, dst, E, s, d);

  float attn[H];
#pragma unroll
  for (int h = 0; h < H; ++h) {
    float al = leaky(a_src[s * H + h] + a_dst[d * H + h]);
    float ea = __expf(al - dec_f32(amax[d * H + h]));
    attn[h] = ea / (denom[d * H + h] + 1e-16f) * 0.125f;  // fold head mean
  }

  const float4* xr = (const float4*)(xp + (size_t)s * 256);
  float* acc = accum + (size_t)d * C;
#pragma unroll
  for (int c4 = 0; c4 < C / 4; ++c4) {
    float4 m4 = make_float4(0.f, 0.f, 0.f, 0.f);
#pragma unroll
    for (int h = 0; h < H; ++h) {
      float4 xv = xr[h * (C / 4) + c4];
      float w = attn[h];
      m4.x += xv.x * w; m4.y += xv.y * w;
      m4.z += xv.z * w; m4.w += xv.w * w;
    }
    float* ap = acc + c4 * 4;
    __hip_atomic_fetch_add(ap + 0, m4.x, __ATOMIC_RELAXED, __HIP_MEMORY_SCOPE_AGENT);
    __hip_atomic_fetch_add(ap + 1, m4.y, __ATOMIC_RELAXED, __HIP_MEMORY_SCOPE_AGENT);
    __hip_atomic_fetch_add(ap + 2, m4.z, __ATOMIC_RELAXED, __HIP_MEMORY_SCOPE_AGENT);
    __hip_atomic_fetch_add(ap + 3, m4.w, __ATOMIC_RELAXED, __HIP_MEMORY_SCOPE_AGENT);
  }
}

// ---------------------------------------------------------------------------
// K6: h = LayerNorm(gat_out + bias_gat + x) with (g1, be1); one thread/node.
// ---------------------------------------------------------------------------
__global__ void k_ln1(const float* __restrict__ accum,
                      const float* __restrict__ x,
                      const float* __restrict__ bias,
                      const float* __restrict__ g1,
                      const float* __restrict__ be1,
                      float* __restrict__ hout, int n) {
  int nd = blockIdx.x * blockDim.x + threadIdx.x;
  if (nd >= n) return;
  float v[C];
  float mu = 0.0f;
#pragma unroll
  for (int c = 0; c < C; ++c) {
    v[c] = accum[(size_t)nd * C + c] + bias[c] + x[(size_t)nd * C + c];
    mu += v[c];
  }
  mu *= (1.0f / C);
  float var = 0.0f;
#pragma unroll
  for (int c = 0; c < C; ++c) {
    float dlt = v[c] - mu;
    var += dlt * dlt;
  }
  float rstd = rsqrtf(var * (1.0f / C) + LN_EPS);
#pragma unroll
  for (int c = 0; c < C; ++c)
    hout[(size_t)nd * C + c] = (v[c] - mu) * rstd * g1[c] + be1[c];
}

// ---------------------------------------------------------------------------
// K7: FFN + residual + LayerNorm, fused, FP32 WMMA.
//   mid = relu(h @ w1 + b1)          (16x32)*(32x128)   8 tiles x 8 ksteps
//   ff  = mid @ w2 + b2              (16x128)*(128x32)  2 tiles x 32 ksteps
//   out = LN(ff + h; g2, be2)
// Weights staged via TDM; per-wave 16-node tile; 16x128 intermediate staged
// through wave-private LDS (same-wave DS ordering + explicit s_wait_dscnt).
// ---------------------------------------------------------------------------
__global__ void __launch_bounds__(256) k_ffn(
    const float* __restrict__ h, const float* __restrict__ w1,
    const float* __restrict__ b1, const float* __restrict__ w2,
    const float* __restrict__ b2, const float* __restrict__ g2,
    const float* __restrict__ be2, float* __restrict__ out, int n) {
  __shared__ float W1[32 * 128];      // 16 KB
  __shared__ float W2[128 * 32];      // 16 KB
  __shared__ float Mid[8][16 * 128];  // 64 KB (8 KB per wave)
#ifdef USE_TDM
  if (threadIdx.x < 32) {
    tdm_load_to_lds(w1, W1, 32 * 128);
    tdm_load_to_lds(w2, W2, 128 * 32);
    __builtin_amdgcn_s_wait_tensorcnt(0);
  }
#else
  for (int i = threadIdx.x; i < 32 * 128; i += blockDim.x) W1[i] = w1[i];
  for (int i = threadIdx.x; i < 128 * 32; i += blockDim.x) W2[i] = w2[i];
#endif
  __syncthreads();

  const int wave = threadIdx.x >> 5;
  const int lane = threadIdx.x & 31;
  const int node0 = (blockIdx.x * 8 + wave) * 16;
  if (node0 >= n) return;

  float* mid = Mid[wave];
  const int row = lane & 15;
  const int khalf = (lane >> 4) * 2;
  const int col = lane & 15;
  const int half = lane >> 4;

  // A fragments for GEMM1 (K = 32 -> 8 ksteps).
  v2f a[8];
  const float* hr = h + (size_t)(node0 + row) * C;
#pragma unroll
  for (int kk = 0; kk < 8; ++kk) {
    a[kk].x = hr[kk * 4 + khalf];
    a[kk].y = hr[kk * 4 + khalf + 1];
  }

  // GEMM1 + bias + ReLU -> LDS intermediate (row-major 16x128).
#pragma unroll 1
  for (int t = 0; t < 8; ++t) {
    v8f c = {};
#pragma unroll
    for (int kk = 0; kk < 8; ++kk) {
      int kr = kk * 4 + khalf;
      v2f b;
      b.x = W1[kr * 128 + t * 16 + col];
      b.y = W1[(kr + 1) * 128 + t * 16 + col];
      c = wmma_f32(a[kk], b, c);
    }
    float bias1 = b1[t * 16 + col];
#pragma unroll
    for (int v = 0; v < 8; ++v) {
      int m = v + half * 8;
      float val = c[v] + bias1;
      mid[m * 128 + t * 16 + col] = val > 0.0f ? val : 0.0f;
    }
  }
  asm volatile("s_wait_dscnt 0" ::: "memory");  // same-wave LDS RAW fence

  // GEMM2 (K = 128 -> 32 ksteps, 2 col tiles of 16).
  v8f c2[2];
#pragma unroll 1
  for (int t = 0; t < 2; ++t) {
    v8f c = {};
#pragma unroll
    for (int kk = 0; kk < 32; ++kk) {
      int kb = kk * 4 + khalf;
      v2f aa, bb;
      aa.x = mid[row * 128 + kb];
      aa.y = mid[row * 128 + kb + 1];
      bb.x = W2[kb * 32 + t * 16 + col];
      bb.y = W2[(kb + 1) * 32 + t * 16 + col];
      c = wmma_f32(aa, bb, c);
    }
    c2[t] = c;
  }

  // Stage pre-LN (ff + b2 + residual h) into LDS, reusing the mid region.
#pragma unroll
  for (int t = 0; t < 2; ++t) {
    int cc = t * 16 + col;
    float bias2 = b2[cc];
#pragma unroll
    for (int v = 0; v < 8; ++v) {
      int m = v + half * 8;
      mid[m * C + cc] = c2[t][v] + bias2 + h[(size_t)(node0 + m) * C + cc];
    }
  }
  asm volatile("s_wait_dscnt 0" ::: "memory");

  // LayerNorm: lanes 0..15 each own one row.
  if (lane < 16) {
    const float* r = mid + lane * C;
    float mu = 0.0f;
#pragma unroll
    for (int c = 0; c < C; ++c) mu += r[c];
    mu *= (1.0f / C);
    float var = 0.0f;
#pragma unroll
    for (int c = 0; c < C; ++c) {
      float dlt = r[c] - mu;
      var += dlt * dlt;
    }
    float rstd = rsqrtf(var * (1.0f / C) + LN_EPS);
    float* o = out + (size_t)(node0 + lane) * C;
#pragma unroll
    for (int c = 0; c < C; ++c) o[c] = (r[c] - mu) * rstd * g2[c] + be2[c];
  }
}

// ---------------------------------------------------------------------------
extern "C" void kernel_launch(void* const* d_in, const int* in_sizes, int n_in,
                              void* d_out, int out_size, void* d_ws,
                              size_t ws_size, hipStream_t stream) {
  const float* x        = (const float*)d_in[0];
  const long long* ei   = (const long long*)d_in[1];  // int64 [2, E]
  const float* W_gat    = (const float*)d_in[2];
  const float* att_src  = (const float*)d_in[3];
  const float* att_dst  = (const float*)d_in[4];
  const float* bias_gat = (const float*)d_in[5];
  const float* w1       = (const float*)d_in[6];
  const float* b1       = (const float*)d_in[7];
  const float* w2       = (const float*)d_in[8];
  const float* b2       = (const float*)d_in[9];
  const float* g1       = (const float*)d_in[10];
  const float* be1      = (const float*)d_in[11];
  const float* g2       = (const float*)d_in[12];
  const float* be2      = (const float*)d_in[13];

  const int N = in_sizes[0] / C;
  const int E = in_sizes[1] / 2;
  const long long* esrc = ei;
  const long long* edst = ei + E;

  // Workspace layout (floats): xp | a_src | a_dst | amax(u32) | denom | accum | h
  float* ws       = (float*)d_ws;
  float* xp       = ws;                               // N*256
  float* asrc     = xp + (size_t)N * 256;             // N*8
  float* adst     = asrc + (size_t)N * H;             // N*8
  unsigned* amax  = (unsigned*)(adst + (size_t)N * H);// N*8
  float* denom    = (float*)(amax + (size_t)N * H);   // N*8
  float* accum    = denom + (size_t)N * H;            // N*32
  float* hbuf     = accum + (size_t)N * C;            // N*32

  const int EN = E + N;

  k_init<<<(N * C + 255) / 256, 256, 0, stream>>>(amax, denom, accum, N);
  k_gemm_xp<<<(N + 127) / 128, 256, 0, stream>>>(x, W_gat, xp, N);
  k_att<<<(N * H + 255) / 256, 256, 0, stream>>>(xp, att_src, att_dst, asrc,
                                                 adst, N);
  k_edge_max<<<(EN + 255) / 256, 256, 0, stream>>>(esrc, edst, asrc, adst,
                                                   amax, E, N);
  k_edge_sum<<<(EN + 255) / 256, 256, 0, stream>>>(esrc, edst, asrc, adst,
                                                   amax, denom, E, N);
  k_edge_agg<<<(EN + 255) / 256, 256, 0, stream>>>(esrc, edst, asrc, adst,
                                                   amax, denom, xp, accum, E, N);
  k_ln1<<<(N + 255) / 256, 256, 0, stream>>>(accum, x, bias_gat, g1, be1,
                                             hbuf, N);
  k_ffn<<<(N + 127) / 128, 256, 0, stream>>>(hbuf, w1, b1, w2, b2, g2, be2,
                                             (float*)d_out, N);
}